// GraphAttentionLayer_52965536694517
// MI455X (gfx1250) — compile-verified
//
#include <hip/hip_runtime.h>
#include <hip/hip_bf16.h>
#include <math.h>

#define D_FEAT 128

typedef __attribute__((ext_vector_type(2))) float v2f;
typedef __attribute__((ext_vector_type(8))) float v8f;

// ---------------------------------------------------------------------------
// order-preserving float <-> uint mapping for atomicMax on floats
// ---------------------------------------------------------------------------
__device__ __forceinline__ unsigned f2o(float f) {
    unsigned u = __float_as_uint(f);
    return (u & 0x80000000u) ? ~u : (u | 0x80000000u);
}
__device__ __forceinline__ float o2f(unsigned u) {
    return (u & 0x80000000u) ? __uint_as_float(u & 0x7FFFFFFFu)
                             : __uint_as_float(~u);
}

__device__ __forceinline__ float edge_logit(const float* __restrict__ s,
                                            const float* __restrict__ t,
                                            const int* __restrict__ row,
                                            const int* __restrict__ col,
                                            int j) {
    float e = s[row[j]] + t[col[j]];
    return e > 0.0f ? e : 0.2f * e;   // leaky_relu, slope 0.2
}

// ---------------------------------------------------------------------------
// Kernel 0: zero output, init reduction scalars
// ---------------------------------------------------------------------------
__global__ void k_init(float4* __restrict__ out4, long long n4,
                       unsigned* __restrict__ gmax, float* __restrict__ gsum) {
    long long i = (long long)blockIdx.x * blockDim.x + threadIdx.x;
    if (i < n4) out4[i] = make_float4(0.f, 0.f, 0.f, 0.f);
    if (i == 0) { *gmax = 0u; *gsum = 0.0f; }   // 0u <= f2o(x) for all x
}

// ---------------------------------------------------------------------------
// Kernel 1: s = h @ att[:128], t = h @ att[128:] via V_WMMA_F32_16X16X4_F32.
// One wave (32 lanes) per 16-row tile of h. B holds att in N-columns 0 and 1.
// A layout (16x4 f32): lanes 0-15 -> M=lane; vgpr0: K=2*half, vgpr1: K=2*half+1.
// B layout (4x16 f32): lanes 0-15 -> N=lane; same K striping as A.
// C/D layout: vgpr v, lanes 0-15 -> M=v, N=lane; lanes 16-31 -> M=v+8, N=lane-16.
// B is built branch-free: load both att halves, blend with per-lane masks.
// ---------------------------------------------------------------------------
__global__ void k_gemv_wmma(const float* __restrict__ h,
                            const float* __restrict__ att,
                            float* __restrict__ s, float* __restrict__ t) {
    const int lane = threadIdx.x;        // 0..31, one wave per block
    const int half = lane >> 4;          // K-half selector
    const int sub  = lane & 15;          // M (for A) / N (for B,C)
    const long long rowA = (long long)blockIdx.x * 16 + sub;
    const float* __restrict__ hrow = h + rowA * D_FEAT;

    const float m0 = (sub == 0) ? 1.0f : 0.0f;   // lane owns N=0 (s column)
    const float m1 = (sub == 1) ? 1.0f : 0.0f;   // lane owns N=1 (t column)

    v8f c = {};
    #pragma unroll 8
    for (int k = 0; k < D_FEAT / 4; ++k) {
        const int kk = 4 * k + 2 * half;
        v2f a;
        a.x = hrow[kk];
        a.y = hrow[kk + 1];
        // unconditional loads of both att halves (contiguous, cache-hot),
        // blended with lane masks -> no exec-mask divergence in the loop
        const float p0 = att[kk];
        const float p1 = att[kk + 1];
        const float q0 = att[D_FEAT + kk];
        const float q1 = att[D_FEAT + kk + 1];
        v2f b;
        b.x = m0 * p0 + m1 * q0;
        b.y = m0 * p1 + m1 * q1;
        c = __builtin_amdgcn_wmma_f32_16x16x4_f32(
                /*neg_a=*/false, a, /*neg_b=*/false, b,
                /*c_mod=*/(short)0, c, /*reuse_a=*/false, /*reuse_b=*/false);
    }

    const long long rbase = (long long)blockIdx.x * 16 + 8 * half;
    if (sub == 0) {               // N=0 column -> s
        #pragma unroll
        for (int v = 0; v < 8; ++v) s[rbase + v] = c[v];
    } else if (sub == 1) {        // N=1 column -> t
        #pragma unroll
        for (int v = 0; v < 8; ++v) t[rbase + v] = c[v];
    }
}

// ---------------------------------------------------------------------------
// Kernel 2: global max of leaky-relu'd edge logits
// ---------------------------------------------------------------------------
__global__ void k_edge_max(const float* __restrict__ s, const float* __restrict__ t,
                           const int* __restrict__ row, const int* __restrict__ col,
                           unsigned* __restrict__ gmax, int E) {
    __shared__ float red[256];
    int tid = threadIdx.x;
    int j = blockIdx.x * blockDim.x + tid;
    float e = (j < E) ? edge_logit(s, t, row, col, j) : -INFINITY;
    red[tid] = e;
    __syncthreads();
    for (int w = 128; w > 0; w >>= 1) {
        if (tid < w) red[tid] = fmaxf(red[tid], red[tid + w]);
        __syncthreads();
    }
    if (tid == 0) atomicMax(gmax, f2o(red[0]));
}

// ---------------------------------------------------------------------------
// Kernel 3: global sum of exp(e - max)
// ---------------------------------------------------------------------------
__global__ void k_edge_sum(const float* __restrict__ s, const float* __restrict__ t,
                           const int* __restrict__ row, const int* __restrict__ col,
                           const unsigned* __restrict__ gmax,
                           float* __restrict__ gsum, int E) {
    __shared__ float red[256];
    int tid = threadIdx.x;
    int j = blockIdx.x * blockDim.x + tid;
    float mx = o2f(*gmax);
    float v = 0.0f;
    if (j < E) v = expf(edge_logit(s, t, row, col, j) - mx);
    red[tid] = v;
    __syncthreads();
    for (int w = 128; w > 0; w >>= 1) {
        if (tid < w) red[tid] += red[tid + w];
        __syncthreads();
    }
    if (tid == 0) atomicAdd(gsum, red[0]);
}

// ---------------------------------------------------------------------------
// Kernel 4: weighted scatter out[row] += (adj * softmax) * h[col].
// 32 lanes per edge, float4 gather of h[col] (global_load_b128),
// 4 atomic f32 adds per lane. h and out are L2-resident (25.6 MB each,
// global L2 = 192 MB), so gathers and atomics resolve at L2, not HBM.
// ---------------------------------------------------------------------------
__global__ void k_scatter(const float* __restrict__ h,
                          const float* __restrict__ adj,
                          const int* __restrict__ row, const int* __restrict__ col,
                          const float* __restrict__ s, const float* __restrict__ t,
                          const unsigned* __restrict__ gmax,
                          const float* __restrict__ gsum,
                          float* __restrict__ out, int E) {
    const int lid  = threadIdx.x & 31;
    const int eidx = blockIdx.x * (blockDim.x >> 5) + (threadIdx.x >> 5);
    if (eidx >= E) return;

    const float mx  = o2f(*gmax);
    const float inv = 1.0f / *gsum;
    const float e   = edge_logit(s, t, row, col, eidx);
    const float w   = adj[eidx] * expf(e - mx) * inv;

    const long long r = row[eidx];
    const long long c = col[eidx];
    const float4* __restrict__ hc = (const float4*)(h + c * D_FEAT);
    float4 v = hc[lid];
    float* o = out + r * D_FEAT + lid * 4;
    atomicAdd(o + 0, w * v.x);
    atomicAdd(o + 1, w * v.y);
    atomicAdd(o + 2, w * v.z);
    atomicAdd(o + 3, w * v.w);
}

// ---------------------------------------------------------------------------
extern "C" void kernel_launch(void* const* d_in, const int* in_sizes, int n_in,
                              void* d_out, int out_size, void* d_ws, size_t ws_size,
                              hipStream_t stream) {
    const float* h    = (const float*)d_in[0];   // [N,128]
    const float* att  = (const float*)d_in[1];   // [256]
    const float* adj  = (const float*)d_in[2];   // [E]
    const int*   row  = (const int*)  d_in[3];   // [E]
    const int*   col  = (const int*)  d_in[4];   // [E]
    float* out = (float*)d_out;                  // [N,128]

    const int N = in_sizes[0] / D_FEAT;          // 50000
    const int E = in_sizes[2];                   // 800000

    // workspace: s[N] | t[N] | gmax (uint) | gsum (float)
    float*    ws_s  = (float*)d_ws;
    float*    ws_t  = ws_s + N;
    unsigned* gmax  = (unsigned*)(ws_t + N);
    float*    gsum  = (float*)(gmax + 1);

    // 0) zero output + init reduction scalars
    {
        long long n4 = (long long)N * D_FEAT / 4;
        int grid = (int)((n4 + 255) / 256);
        k_init<<<grid, 256, 0, stream>>>((float4*)out, n4, gmax, gsum);
    }
    // 1) s,t GEMV via WMMA (one wave per 16 rows; N = 50000 = 3125 * 16)
    {
        int grid = (N + 15) / 16;
        k_gemv_wmma<<<grid, 32, 0, stream>>>(h, att, ws_s, ws_t);
    }
    // 2) global max over edge logits
    {
        int grid = (E + 255) / 256;
        k_edge_max<<<grid, 256, 0, stream>>>(ws_s, ws_t, row, col, gmax, E);
    }
    // 3) global sum of exp
    {
        int grid = (E + 255) / 256;
        k_edge_sum<<<grid, 256, 0, stream>>>(ws_s, ws_t, row, col, gmax, gsum, E);
    }
    // 4) softmax-weighted scatter-add
    {
        int edgesPerBlock = 256 / 32;
        int grid = (E + edgesPerBlock - 1) / edgesPerBlock;
        k_scatter<<<grid, 256, 0, stream>>>(h, adj, row, col, ws_s, ws_t,
                                            gmax, gsum, out, E);
    }
}